// FlattenLoss_1434519077376
// MI455X (gfx1250) — compile-verified
//
#include <hip/hip_runtime.h>

#define FL_EPS 1e-6f
#define BLOCK 256

#if defined(__HIP_DEVICE_COMPILE__) && __has_builtin(__builtin_amdgcn_global_load_async_to_lds_b32)
#define FL_HAS_ASYNC 1
#else
#define FL_HAS_ASYNC 0
#endif

#if FL_HAS_ASYNC
typedef __attribute__((address_space(1))) int fl_gint;
typedef __attribute__((address_space(3))) int fl_lint;
#endif

// ---------------------------------------------------------------------------
// Pass 1: per (edge-tile, batch-group) block.  Each thread owns one edge,
// loads its 4 vertex indices ONCE (staged through LDS via the CDNA5 async
// global->LDS path when available), then loops over NB batches' vertex slabs
// accumulating the flatten-loss term for each batch.
// ---------------------------------------------------------------------------
template <int NB>
__global__ __launch_bounds__(BLOCK) void fl_edge_kernel(
    const float* __restrict__ vertices,   // (B, V, 3)
    const int*   __restrict__ v0s,
    const int*   __restrict__ v1s,
    const int*   __restrict__ v2s,
    const int*   __restrict__ v3s,
    float*       __restrict__ partials,   // (B, nblk)
    int V, int E)
{
    const int tid = threadIdx.x;
    const int e   = blockIdx.x * BLOCK + tid;
    const int e_c = (e < E) ? e : (E - 1);          // clamp; mask at the end
    const float validf = (e < E) ? 1.0f : 0.0f;
    const int b0 = blockIdx.y * NB;

    // ---- stage this tile's indices (async global -> LDS on CDNA5) ----
    __shared__ int idx_lds[4 * BLOCK];
#if FL_HAS_ASYNC
    __builtin_amdgcn_global_load_async_to_lds_b32(
        (fl_gint*)(v0s + e_c), (fl_lint*)&idx_lds[0 * BLOCK + tid], 0, 0);
    __builtin_amdgcn_global_load_async_to_lds_b32(
        (fl_gint*)(v1s + e_c), (fl_lint*)&idx_lds[1 * BLOCK + tid], 0, 0);
    __builtin_amdgcn_global_load_async_to_lds_b32(
        (fl_gint*)(v2s + e_c), (fl_lint*)&idx_lds[2 * BLOCK + tid], 0, 0);
    __builtin_amdgcn_global_load_async_to_lds_b32(
        (fl_gint*)(v3s + e_c), (fl_lint*)&idx_lds[3 * BLOCK + tid], 0, 0);
#if __has_builtin(__builtin_amdgcn_s_wait_asynccnt)
    __builtin_amdgcn_s_wait_asynccnt(0);
#else
    asm volatile("s_wait_asynccnt 0x0" ::: "memory");
#endif
    const int i0 = idx_lds[0 * BLOCK + tid];
    const int i1 = idx_lds[1 * BLOCK + tid];
    const int i2 = idx_lds[2 * BLOCK + tid];
    const int i3 = idx_lds[3 * BLOCK + tid];
#else
    const int i0 = v0s[e_c];
    const int i1 = v1s[e_c];
    const int i2 = v2s[e_c];
    const int i3 = v3s[e_c];
    (void)idx_lds;
#endif

    float acc[NB];
#pragma unroll
    for (int bb = 0; bb < NB; ++bb) acc[bb] = 0.0f;

#pragma unroll
    for (int bb = 0; bb < NB; ++bb) {
        const size_t base = (size_t)(b0 + bb) * (size_t)V * 3u;
        const float* __restrict__ q0 = vertices + base + (size_t)i0 * 3u;
        const float* __restrict__ q1 = vertices + base + (size_t)i1 * 3u;
        const float* __restrict__ q2 = vertices + base + (size_t)i2 * 3u;
        const float* __restrict__ q3 = vertices + base + (size_t)i3 * 3u;

        const float p0x = q0[0], p0y = q0[1], p0z = q0[2];
        const float p1x = q1[0], p1y = q1[1], p1z = q1[2];
        const float p2x = q2[0], p2y = q2[1], p2z = q2[2];
        const float p3x = q3[0], p3y = q3[1], p3z = q3[2];

        // a = p1 - p0
        const float ax = p1x - p0x, ay = p1y - p0y, az = p1z - p0z;
        const float al2 = ax*ax + ay*ay + az*az;
        const float al1 = sqrtf(al2 + FL_EPS);
        const float inv_al2e = 1.0f / (al2 + FL_EPS);

        // wing 1: b1 = p2 - p0
        const float b1x = p2x - p0x, b1y = p2y - p0y, b1z = p2z - p0z;
        const float b1l2 = b1x*b1x + b1y*b1y + b1z*b1z;
        const float b1l1 = sqrtf(b1l2 + FL_EPS);
        const float ab1  = ax*b1x + ay*b1y + az*b1z;
        const float cos1 = ab1 / (al1 * b1l1 + FL_EPS);
        const float sin1 = sqrtf(1.0f - cos1*cos1 + FL_EPS);
        const float t1   = ab1 * inv_al2e;
        const float cb1x = b1x - ax*t1, cb1y = b1y - ay*t1, cb1z = b1z - az*t1;
        const float cb1l1 = b1l1 * sin1;

        // wing 2: b2 = p3 - p0
        const float b2x = p3x - p0x, b2y = p3y - p0y, b2z = p3z - p0z;
        const float b2l2 = b2x*b2x + b2y*b2y + b2z*b2z;
        const float b2l1 = sqrtf(b2l2 + FL_EPS);
        const float ab2  = ax*b2x + ay*b2y + az*b2z;
        const float cos2 = ab2 / (al1 * b2l1 + FL_EPS);
        const float sin2 = sqrtf(1.0f - cos2*cos2 + FL_EPS);
        const float t2   = ab2 * inv_al2e;
        const float cb2x = b2x - ax*t2, cb2y = b2y - ay*t2, cb2z = b2z - az*t2;
        const float cb2l1 = b2l1 * sin2;

        const float num = cb1x*cb2x + cb1y*cb2y + cb1z*cb2z;
        const float c   = num / (cb1l1 * cb2l1 + FL_EPS);
        const float u   = c + 1.0f;
        acc[bb] = u * u * validf;
    }

    // ---- block reduction: wave32 butterfly per batch, then cross-wave ----
    __shared__ float wsum[BLOCK / 32][NB];
    const int lane = tid & 31;
    const int wid  = tid >> 5;

#pragma unroll
    for (int bb = 0; bb < NB; ++bb) {
        float s = acc[bb];
#pragma unroll
        for (int off = 16; off > 0; off >>= 1)
            s += __shfl_xor(s, off, 32);
        if (lane == 0) wsum[wid][bb] = s;
    }
    __syncthreads();

    if (tid < NB) {
        float t = 0.0f;
#pragma unroll
        for (int w = 0; w < BLOCK / 32; ++w) t += wsum[w][tid];
        partials[(size_t)(b0 + tid) * gridDim.x + blockIdx.x] = t;
    }
}

// ---------------------------------------------------------------------------
// Pass 2: deterministic reduction of per-block partials into out[b].
// ---------------------------------------------------------------------------
__global__ __launch_bounds__(BLOCK) void fl_reduce_kernel(
    const float* __restrict__ partials,   // (B, nblk)
    float*       __restrict__ out,        // (B,)
    int nblk)
{
    const int b = blockIdx.x;
    float s = 0.0f;
    for (int i = threadIdx.x; i < nblk; i += BLOCK)
        s += partials[(size_t)b * nblk + i];

#pragma unroll
    for (int off = 16; off > 0; off >>= 1)
        s += __shfl_xor(s, off, 32);

    __shared__ float warp_acc[BLOCK / 32];
    const int lane = threadIdx.x & 31;
    const int wid  = threadIdx.x >> 5;
    if (lane == 0) warp_acc[wid] = s;
    __syncthreads();

    if (threadIdx.x == 0) {
        float t = 0.0f;
#pragma unroll
        for (int w = 0; w < BLOCK / 32; ++w) t += warp_acc[w];
        out[b] = t;
    }
}

extern "C" void kernel_launch(void* const* d_in, const int* in_sizes, int n_in,
                              void* d_out, int out_size, void* d_ws, size_t ws_size,
                              hipStream_t stream) {
    const float* vertices = (const float*)d_in[0];
    const int*   v0s      = (const int*)d_in[1];
    const int*   v1s      = (const int*)d_in[2];
    const int*   v2s      = (const int*)d_in[3];
    const int*   v3s      = (const int*)d_in[4];
    float*       out      = (float*)d_out;

    const int B = out_size;                                           // 64
    const int E = in_sizes[1];                                        // 300000
    const int V = (int)((long long)in_sizes[0] / ((long long)B * 3)); // 100000

    const int nblk = (E + BLOCK - 1) / BLOCK;     // edge tiles
    float* partials = (float*)d_ws;               // needs B*nblk*4 bytes (~300 KB)

    if ((B & 7) == 0) {
        dim3 grid(nblk, B / 8);
        fl_edge_kernel<8><<<grid, BLOCK, 0, stream>>>(vertices, v0s, v1s, v2s, v3s,
                                                      partials, V, E);
    } else {
        dim3 grid(nblk, B);
        fl_edge_kernel<1><<<grid, BLOCK, 0, stream>>>(vertices, v0s, v1s, v2s, v3s,
                                                      partials, V, E);
    }

    fl_reduce_kernel<<<B, BLOCK, 0, stream>>>(partials, out, nblk);
}